// AGNN_39041252720981
// MI455X (gfx1250) — compile-verified
//
#include <hip/hip_runtime.h>
#include <math.h>

typedef __attribute__((ext_vector_type(2))) float v2f;
typedef __attribute__((ext_vector_type(8))) float v8f;

#define EPSV 1e-12f

// ---------- helpers ----------
__device__ __forceinline__ v8f wmma_f32_4(v2f a, v2f b, v8f c) {
  // V_WMMA_F32_16X16X4_F32 : D(16x16,f32) = A(16x4) * B(4x16) + C
  return __builtin_amdgcn_wmma_f32_16x16x4_f32(
      /*neg_a=*/false, a, /*neg_b=*/false, b,
      /*c_mod=*/(short)0, c, /*reuse_a=*/false, /*reuse_b=*/false);
}

__device__ __forceinline__ float wave_sum32(float v) {
  #pragma unroll
  for (int off = 16; off > 0; off >>= 1) v += __shfl_xor(v, off, 32);
  return v;
}
__device__ __forceinline__ float wave_max32(float v) {
  #pragma unroll
  for (int off = 16; off > 0; off >>= 1) v = fmaxf(v, __shfl_xor(v, off, 32));
  return v;
}

// Monotonic float->uint key so unsigned atomicMax == float max.
__device__ __forceinline__ unsigned fkey(float x) {
  unsigned u = __float_as_uint(x);
  return (u & 0x80000000u) ? ~u : (u | 0x80000000u);
}
__device__ __forceinline__ float funkey(unsigned k) {
  return __uint_as_float((k & 0x80000000u) ? (k & 0x7FFFFFFFu) : ~k);
}
#define KEY_NEG_INF 0x007FFFFFu   // fkey(-inf)

// ---------- GEMM1: H = relu(X[M,128] @ W1[32,128]^T + b1), via f32 WMMA ----------
__global__ __launch_bounds__(256)
void k_gemm1_relu(const float* __restrict__ X, const float* __restrict__ W,
                  const float* __restrict__ bias, float* __restrict__ H, int M) {
  int wave = (int)((blockIdx.x * blockDim.x + threadIdx.x) >> 5);
  int lane = threadIdx.x & 31;
  int m0 = wave * 16;
  if (m0 >= M) return;
  int half = lane >> 4;          // which half-wave
  int lid  = lane & 15;
  int arow = m0 + lid; if (arow >= M) arow = M - 1;   // clamp for partial tail
  const float* ap = X + (size_t)arow * 128;
  bool full = (m0 + 16 <= M);    // wave-uniform

  #pragma unroll
  for (int nt = 0; nt < 2; ++nt) {
    int n0 = nt * 16;
    const float* bp = W + (size_t)(n0 + lid) * 128;   // B[k][n] = W[n][k]
    v8f acc = {0.f,0.f,0.f,0.f,0.f,0.f,0.f,0.f};
    #pragma unroll 4
    for (int k = 0; k < 128; k += 4) {
      int k0 = k + half * 2;                 // lanes 0-15: K,K+1 ; lanes 16-31: K+2,K+3
      v2f a = { ap[k0], ap[k0 + 1] };
      v2f b = { bp[k0], bp[k0 + 1] };
      acc = wmma_f32_4(a, b, acc);
    }
    float bv = bias[n0 + lid];
    float* hp = H + (size_t)(m0 + half * 8) * 32 + n0 + lid;
    if (full) {
      #pragma unroll
      for (int r = 0; r < 8; ++r)
        hp[(size_t)r * 32] = fmaxf(acc[r] + bv, 0.f);
    } else {
      #pragma unroll
      for (int r = 0; r < 8; ++r) {
        int m = m0 + r + half * 8;
        if (m < M) H[(size_t)m * 32 + n0 + lid] = fmaxf(acc[r] + bv, 0.f);
      }
    }
  }
}

// ---------- GEMM2: OUT = H[M,32] @ W2[64,32]^T + b2 (pre-softmax logits) ----------
__global__ __launch_bounds__(256)
void k_gemm2(const float* __restrict__ H, const float* __restrict__ W,
             const float* __restrict__ bias, float* __restrict__ OUT, int M) {
  int wave = (int)((blockIdx.x * blockDim.x + threadIdx.x) >> 5);
  int lane = threadIdx.x & 31;
  int m0 = wave * 16;
  if (m0 >= M) return;
  int half = lane >> 4;
  int lid  = lane & 15;
  int arow = m0 + lid; if (arow >= M) arow = M - 1;
  const float* ap = H + (size_t)arow * 32;
  bool full = (m0 + 16 <= M);

  #pragma unroll
  for (int nt = 0; nt < 4; ++nt) {
    int n0 = nt * 16;
    const float* bp = W + (size_t)(n0 + lid) * 32;
    v8f acc = {0.f,0.f,0.f,0.f,0.f,0.f,0.f,0.f};
    #pragma unroll
    for (int k = 0; k < 32; k += 4) {
      int k0 = k + half * 2;
      v2f a = { ap[k0], ap[k0 + 1] };
      v2f b = { bp[k0], bp[k0 + 1] };
      acc = wmma_f32_4(a, b, acc);
    }
    float bv = bias[n0 + lid];
    float* op = OUT + (size_t)(m0 + half * 8) * 64 + n0 + lid;
    if (full) {
      #pragma unroll
      for (int r = 0; r < 8; ++r)
        op[(size_t)r * 64] = acc[r] + bv;
    } else {
      #pragma unroll
      for (int r = 0; r < 8; ++r) {
        int m = m0 + r + half * 8;
        if (m < M) OUT[(size_t)m * 64 + n0 + lid] = acc[r] + bv;
      }
    }
  }
}

// ---------- Per-node prep: HN = H/max(||H||,eps); zero accumulators ----------
__global__ __launch_bounds__(256)
void k_rownorm_prep(const float* __restrict__ H, float* __restrict__ HN,
                    unsigned* __restrict__ MKEY, float* __restrict__ DEN,
                    float* __restrict__ OUTACC, int N) {
  int node = (int)((blockIdx.x * blockDim.x + threadIdx.x) >> 5);
  int lane = threadIdx.x & 31;
  if (node >= N) return;
  float v = H[(size_t)node * 32 + lane];
  float s = wave_sum32(v * v);
  float inv = 1.f / fmaxf(sqrtf(s), EPSV);
  HN[(size_t)node * 32 + lane] = v * inv;
  OUTACC[(size_t)node * 32 + lane] = 0.f;
  if (lane == 0) { MKEY[node] = KEY_NEG_INF; DEN[node] = 0.f; }
}

// ---------- Edge scores: 32 edges per wave; e = beta * <hn[src], hn[dst]> ----------
__global__ __launch_bounds__(256)
void k_edge_scores(const float* __restrict__ HN, const int* __restrict__ src,
                   const int* __restrict__ dst, const float* __restrict__ betas,
                   int layer, float* __restrict__ E, unsigned* __restrict__ MKEY,
                   int nE) {
  int wave = (int)((blockIdx.x * blockDim.x + threadIdx.x) >> 5);
  int lane = threadIdx.x & 31;
  int e0 = wave * 32;
  if (e0 >= nE) return;
  int myE = e0 + lane;
  bool has = myE < nE;
  int es = has ? src[myE] : 0;        // coalesced index loads
  int ed = has ? dst[myE] : 0;
  float beta = betas[layer];
  int cnt = min(32, nE - e0);
  float myev = 0.f;

  for (int i = 0; i < cnt; ++i) {
    int s = __shfl(es, i, 32);
    int d = __shfl(ed, i, 32);
    if (i + 1 < cnt) {                 // prefetch next edge's rows (L2 -> L0)
      int s2 = __shfl(es, i + 1, 32);
      int d2 = __shfl(ed, i + 1, 32);
      __builtin_prefetch(&HN[(size_t)s2 * 32 + lane], 0, 3);
      __builtin_prefetch(&HN[(size_t)d2 * 32 + lane], 0, 3);
    }
    float p = HN[(size_t)s * 32 + lane] * HN[(size_t)d * 32 + lane];
    float dotv = wave_sum32(p);        // all lanes hold the dot
    if (i == lane) myev = beta * dotv; // keep own edge's score
  }
  if (has) {
    E[myE] = myev;                     // coalesced score store
    atomicMax(&MKEY[ed], fkey(myev));  // one atomic instruction for 32 edges
  }
}

// ---------- Finalize segment max (zero-in-degree guard) ----------
__global__ __launch_bounds__(256)
void k_finalize_max(const unsigned* __restrict__ MKEY, float* __restrict__ MVAL, int N) {
  int i = (int)(blockIdx.x * blockDim.x + threadIdx.x);
  if (i >= N) return;
  unsigned k = MKEY[i];
  MVAL[i] = (k == KEY_NEG_INF) ? 0.f : funkey(k);
}

// ---------- ex = exp(e - m[dst]); denom[dst] += ex ----------
__global__ __launch_bounds__(256)
void k_edge_exp(float* __restrict__ E, const int* __restrict__ dst,
                const float* __restrict__ MVAL, float* __restrict__ DEN, int nE) {
  int j = (int)(blockIdx.x * blockDim.x + threadIdx.x);
  if (j >= nE) return;
  int d = dst[j];
  float ex = expf(E[j] - MVAL[d]);
  E[j] = ex;
  atomicAdd(&DEN[d], ex);
}

// ---------- 32 edges per wave: out[dst] += alpha * h[src] ----------
__global__ __launch_bounds__(256)
void k_edge_scatter(const float* __restrict__ H, const float* __restrict__ E,
                    const float* __restrict__ DEN, const int* __restrict__ src,
                    const int* __restrict__ dst, float* __restrict__ OUTACC, int nE) {
  int wave = (int)((blockIdx.x * blockDim.x + threadIdx.x) >> 5);
  int lane = threadIdx.x & 31;
  int e0 = wave * 32;
  if (e0 >= nE) return;
  int myE = e0 + lane;
  bool has = myE < nE;
  int es = has ? src[myE] : 0;                       // coalesced
  int ed = has ? dst[myE] : 0;                       // coalesced
  // per-lane alpha for own edge: coalesced E load + one gather of denom
  float myalpha = has ? E[myE] / fmaxf(DEN[ed], EPSV) : 0.f;
  int cnt = min(32, nE - e0);

  for (int i = 0; i < cnt; ++i) {
    int s = __shfl(es, i, 32);
    int d = __shfl(ed, i, 32);
    float alpha = __shfl(myalpha, i, 32);
    if (i + 1 < cnt) {
      int s2 = __shfl(es, i + 1, 32);
      __builtin_prefetch(&H[(size_t)s2 * 32 + lane], 0, 3);
    }
    atomicAdd(&OUTACC[(size_t)d * 32 + lane], alpha * H[(size_t)s * 32 + lane]);
  }
}

// ---------- elementwise ReLU ----------
__global__ __launch_bounds__(256)
void k_relu(float* __restrict__ X, int n) {
  int i = (int)(blockIdx.x * blockDim.x + threadIdx.x);
  if (i < n) X[i] = fmaxf(X[i], 0.f);
}

// ---------- in-place log_softmax over rows of 64 ----------
__global__ __launch_bounds__(256)
void k_logsoftmax64(float* __restrict__ OUT, int M) {
  int row = (int)((blockIdx.x * blockDim.x + threadIdx.x) >> 5);
  int lane = threadIdx.x & 31;
  if (row >= M) return;
  float* p = OUT + (size_t)row * 64;
  float a = p[lane], b = p[lane + 32];
  float mx = wave_max32(fmaxf(a, b));
  float s = wave_sum32(expf(a - mx) + expf(b - mx));
  float lse = mx + logf(s);
  p[lane] = a - lse;
  p[lane + 32] = b - lse;
}

extern "C" void kernel_launch(void* const* d_in, const int* in_sizes, int n_in,
                              void* d_out, int out_size, void* d_ws, size_t ws_size,
                              hipStream_t stream) {
  const float* features = (const float*)d_in[0];
  const int*   src      = (const int*)d_in[1];
  const int*   dst      = (const int*)d_in[2];
  const float* w1       = (const float*)d_in[3];
  const float* b1       = (const float*)d_in[4];
  const float* betas    = (const float*)d_in[5];
  const float* w2       = (const float*)d_in[6];
  const float* b2       = (const float*)d_in[7];
  float* out = (float*)d_out;

  const int N  = in_sizes[0] / 128;  // nodes
  const int En = in_sizes[1];        // edges

  // workspace layout (floats)
  float*    hA   = (float*)d_ws;
  float*    hB   = hA + (size_t)N * 32;
  float*    hn   = hB + (size_t)N * 32;
  float*    Ebuf = hn + (size_t)N * 32;
  unsigned* MKEY = (unsigned*)(Ebuf + (size_t)En);
  float*    MVAL = (float*)(MKEY + N);
  float*    DEN  = MVAL + N;

  const int TB = 256;               // 8 waves per block (wave32)
  const int mtiles = (N + 15) / 16;
  const int waveBlocksM = (mtiles + 7) / 8;
  const int waveBlocksN = (N + 7) / 8;
  const int ewaves = (En + 31) / 32;           // 32 edges per wave
  const int waveBlocksE = (ewaves + 7) / 8;

  // 1) h = relu(features @ w1^T + b1)   [WMMA f32]
  k_gemm1_relu<<<waveBlocksM, TB, 0, stream>>>(features, w1, b1, hA, N);

  // 2) two AGNN conv layers
  float* hin = hA; float* hout = hB;
  for (int layer = 0; layer < 2; ++layer) {
    k_rownorm_prep<<<waveBlocksN, TB, 0, stream>>>(hin, hn, MKEY, DEN, hout, N);
    k_edge_scores <<<waveBlocksE, TB, 0, stream>>>(hn, src, dst, betas, layer, Ebuf, MKEY, En);
    k_finalize_max<<<(N + TB - 1) / TB, TB, 0, stream>>>(MKEY, MVAL, N);
    k_edge_exp    <<<(En + TB - 1) / TB, TB, 0, stream>>>(Ebuf, dst, MVAL, DEN, En);
    k_edge_scatter<<<waveBlocksE, TB, 0, stream>>>(hin, Ebuf, DEN, src, dst, hout, En);
    k_relu        <<<((N * 32) + TB - 1) / TB, TB, 0, stream>>>(hout, N * 32);
    float* t = hin; hin = hout; hout = t;
  }

  // 3) logits = h @ w2^T + b2           [WMMA f32]
  k_gemm2<<<waveBlocksM, TB, 0, stream>>>(hin, w2, b2, out, N);

  // 4) in-place log_softmax over 64 classes
  k_logsoftmax64<<<waveBlocksN, TB, 0, stream>>>(out, N);
}